// MoEBlock_2499670966563
// MI455X (gfx1250) — compile-verified
//
#include <hip/hip_runtime.h>
#include <hip/hip_bf16.h>

// Problem constants (match reference: B=4, S=2048, H=2048, E=8)
constexpr int kB = 4;
constexpr int kS = 2048;
constexpr int kH = 2048;
constexpr int kE = 8;
constexpr int kT = kB * kS;          // 8192 tokens

typedef float v2f __attribute__((ext_vector_type(2)));
typedef float v8f __attribute__((ext_vector_type(8)));

// ---------------------------------------------------------------------------
// Kernel 0: zero counts + cursors in workspace
// ---------------------------------------------------------------------------
__global__ void moe_init_kernel(int* counts, int* cursor) {
    int t = threadIdx.x;
    if (t < kE) { counts[t] = 0; cursor[t] = 0; }
}

// ---------------------------------------------------------------------------
// Kernel 1: gating. One wave32 per token: logits = x[t,:]·Wg (E=8 experts),
// softmax -> top-1 gate prob + expert index. Histogram per-expert counts.
// ---------------------------------------------------------------------------
__global__ __launch_bounds__(256)
void moe_gate_kernel(const float* __restrict__ x, const float* __restrict__ Wg,
                     float* __restrict__ gate, int* __restrict__ eidx,
                     int* __restrict__ counts) {
    const int lane = threadIdx.x & 31;
    const int wave = threadIdx.x >> 5;
    const int tok  = blockIdx.x * 8 + wave;   // 8 waves per block
    if (tok >= kT) return;

    const float* xr = x + (size_t)tok * kH;
    float acc[kE];
#pragma unroll
    for (int e = 0; e < kE; ++e) acc[e] = 0.0f;

    for (int h = lane; h < kH; h += 32) {
        const float xv = xr[h];
        const float4 w0 = ((const float4*)(Wg + (size_t)h * kE))[0];
        const float4 w1 = ((const float4*)(Wg + (size_t)h * kE))[1];
        acc[0] += xv * w0.x; acc[1] += xv * w0.y;
        acc[2] += xv * w0.z; acc[3] += xv * w0.w;
        acc[4] += xv * w1.x; acc[5] += xv * w1.y;
        acc[6] += xv * w1.z; acc[7] += xv * w1.w;
    }
    // wave32 cross-lane reduction
#pragma unroll
    for (int m = 16; m >= 1; m >>= 1) {
#pragma unroll
        for (int e = 0; e < kE; ++e) acc[e] += __shfl_xor(acc[e], m, 32);
    }
    if (lane == 0) {
        float lmax = acc[0]; int im = 0;
#pragma unroll
        for (int e = 1; e < kE; ++e) {
            if (acc[e] > lmax) { lmax = acc[e]; im = e; }
        }
        float s = 0.0f;
#pragma unroll
        for (int e = 0; e < kE; ++e) s += __expf(acc[e] - lmax);
        gate[tok] = 1.0f / s;      // softmax prob of the argmax expert
        eidx[tok] = im;
        atomicAdd(&counts[im], 1);
    }
}

// ---------------------------------------------------------------------------
// Kernel 2: exclusive scan over 8 expert counts (single thread — trivial)
// ---------------------------------------------------------------------------
__global__ void moe_scan_kernel(const int* __restrict__ counts,
                                int* __restrict__ offsets) {
    if (threadIdx.x == 0) {
        int run = 0;
        for (int e = 0; e < kE; ++e) { offsets[e] = run; run += counts[e]; }
        offsets[kE] = run;
    }
}

// ---------------------------------------------------------------------------
// Kernel 3: scatter token ids into expert-sorted permutation
// (intra-expert order is irrelevant: every token row is independent)
// ---------------------------------------------------------------------------
__global__ __launch_bounds__(256)
void moe_scatter_kernel(const int* __restrict__ eidx,
                        const int* __restrict__ offsets,
                        int* __restrict__ cursor, int* __restrict__ perm) {
    const int tok = blockIdx.x * blockDim.x + threadIdx.x;
    if (tok >= kT) return;
    const int e = eidx[tok];
    const int pos = atomicAdd(&cursor[e], 1);
    perm[offsets[e] + pos] = tok;
}

// ---------------------------------------------------------------------------
// Kernel 4: grouped GEMM with V_WMMA_F32_16X16X4_F32.
// grid = (H/128 N-tiles, T/16 token tiles, E experts); block = 128 (4 waves).
// Each block: 16 gathered token rows x 128 output cols. Each wave owns a
// 16x32 output tile (two N strips -> two accumulators), so every A fragment
// read from LDS feeds TWO wmma ops. A tile staged in LDS, B streamed from W.
// ---------------------------------------------------------------------------
constexpr int KC = 64;    // K chunk staged in LDS
constexpr int BN = 128;   // output cols per block (4 waves x 32)

__global__ __launch_bounds__(128)
void moe_gemm_kernel(const float* __restrict__ x, const float* __restrict__ W,
                     const float* __restrict__ bias,
                     const float* __restrict__ gate,
                     const int* __restrict__ perm,
                     const int* __restrict__ offsets,
                     float* __restrict__ out) {
    const int e = blockIdx.z;
    const int segBeg = offsets[e];
    const int segEnd = offsets[e + 1];
    const int tileBeg = segBeg + blockIdx.y * 16;
    if (tileBeg >= segEnd) return;                 // block-uniform early exit
    const int rowsValid = min(16, segEnd - tileBeg);

    __shared__ float As[16][KC + 4];               // +4 pad vs bank conflicts
    __shared__ int   stok[16];
    __shared__ float sgate[16];

    if (threadIdx.x < 16) {
        const int r = threadIdx.x;
        const int t = (r < rowsValid) ? perm[tileBeg + r] : -1;
        stok[r]  = t;
        sgate[r] = (t >= 0) ? gate[t] : 0.0f;
    }

    const int lane   = threadIdx.x & 31;
    const int wid    = threadIdx.x >> 5;
    const int n0     = blockIdx.x * BN + wid * 32; // wave's 32-wide window
    const int nLane0 = n0 + (lane & 15);           // strip 0 column
    const int nLane1 = nLane0 + 16;                // strip 1 column
    const int kHalf  = (lane >> 4) << 1;           // 0 for lanes 0-15, 2 for 16-31
    const int row    = lane & 15;

    const float* We = W + (size_t)e * kH * kH;

    v8f c0 = {};                                   // f32 accumulators
    v8f c1 = {};

    for (int k0 = 0; k0 < kH; k0 += KC) {
        __syncthreads();                           // protect previous A tile
        // Cooperative gather of A tile: 16 rows x KC floats as float4s
        for (int i = threadIdx.x; i < 16 * (KC / 4); i += 128) {
            const int r  = i >> 4;
            const int c4 = i & 15;
            const int t  = stok[r];
            float4 v = make_float4(0.f, 0.f, 0.f, 0.f);
            if (t >= 0)
                v = ((const float4*)(x + (size_t)t * kH + k0))[c4];
            *(float4*)&As[r][c4 * 4] = v;
        }
        __syncthreads();

        // Prefetch next K-chunk of W[e] (global_prefetch_b8 path)
        if (k0 + KC < kH) {
            __builtin_prefetch(&We[(size_t)(k0 + KC + kHalf) * kH + nLane0], 0, 1);
            __builtin_prefetch(&We[(size_t)(k0 + KC + kHalf) * kH + nLane1], 0, 1);
        }

#pragma unroll
        for (int ks = 0; ks < KC / 4; ++ks) {
            const int kl = ks * 4 + kHalf;         // local K within tile
            // A fragment per ISA 16x4 layout: lanes0-15 -> K0,K1; 16-31 -> K2,K3
            v2f a;
            a.x = As[row][kl];
            a.y = As[row][kl + 1];
            // B fragments (mirror layout): W[e][k][n], n coalesced across lanes
            const int kg = k0 + kl;
            const size_t r0 = (size_t)kg * kH;
            const size_t r1 = (size_t)(kg + 1) * kH;
            v2f b0, b1;
            b0.x = We[r0 + nLane0];
            b0.y = We[r1 + nLane0];
            b1.x = We[r0 + nLane1];
            b1.y = We[r1 + nLane1];
            c0 = __builtin_amdgcn_wmma_f32_16x16x4_f32(
                     false, a, false, b0, (short)0, c0, false, false);
            c1 = __builtin_amdgcn_wmma_f32_16x16x4_f32(
                     false, a, false, b1, (short)0, c1, false, false);
        }
    }

    // Epilogue: C/D layout — VGPR v: lanes0-15 -> M=v, lanes16-31 -> M=v+8
    const float bv0  = bias[(size_t)e * kH + nLane0];
    const float bv1  = bias[(size_t)e * kH + nLane1];
    const int   mOff = (lane >> 4) << 3;           // 0 or 8
#pragma unroll
    for (int v = 0; v < 8; ++v) {
        const int m = v + mOff;
        if (m < rowsValid) {
            const int t = stok[m];
            const float g = sgate[m];
            out[(size_t)t * kH + nLane0] = g * (c0[v] + bv0);
            out[(size_t)t * kH + nLane1] = g * (c1[v] + bv1);
        }
    }
}

// ---------------------------------------------------------------------------
// Host-side launcher
// ---------------------------------------------------------------------------
extern "C" void kernel_launch(void* const* d_in, const int* in_sizes, int n_in,
                              void* d_out, int out_size, void* d_ws, size_t ws_size,
                              hipStream_t stream) {
    const float* x    = (const float*)d_in[0];   // [B,S,H]
    const float* Wg   = (const float*)d_in[1];   // [H,E]
    const float* W    = (const float*)d_in[2];   // [E,H,H]
    const float* bias = (const float*)d_in[3];   // [E,H]
    float*       out  = (float*)d_out;           // [B,S,H]

    // Workspace carve-up
    float* gate    = (float*)d_ws;               // kT floats
    int*   eidx    = (int*)(gate + kT);          // kT ints
    int*   counts  = eidx + kT;                  // 16 ints (8 used)
    int*   offsets = counts + 16;                // 16 ints (9 used)
    int*   cursor  = offsets + 16;               // 16 ints (8 used)
    int*   perm    = cursor + 16;                // kT ints

    moe_init_kernel<<<1, 32, 0, stream>>>(counts, cursor);

    moe_gate_kernel<<<kT / 8, 256, 0, stream>>>(x, Wg, gate, eidx, counts);

    moe_scan_kernel<<<1, 32, 0, stream>>>(counts, offsets);

    moe_scatter_kernel<<<kT / 256, 256, 0, stream>>>(eidx, offsets, cursor, perm);

    dim3 grid(kH / BN, kT / 16, kE);             // (16, 512, 8); surplus tiles exit
    moe_gemm_kernel<<<grid, 128, 0, stream>>>(x, W, bias, gate, perm, offsets, out);
}